// GMMQuantizer_35845797053134
// MI455X (gfx1250) — compile-verified
//
#include <hip/hip_runtime.h>
#include <hip/hip_bf16.h>
#include <math.h>

#define LCOMP 16
#define TPB   256
#define MAXBLOCKS 1024

// ---- gfx1250 async-to-LDS plumbing -----------------------------------------
#if defined(__has_builtin)
#  if __has_builtin(__builtin_amdgcn_global_load_async_to_lds_b128)
#    define HAVE_ASYNC_BUILTIN 1
#  endif
#  if __has_builtin(__builtin_amdgcn_s_wait_asynccnt)
#    define HAVE_WAITASYNC_BUILTIN 1
#  endif
#  if __has_builtin(__builtin_amdgcn_exp2f)
#    define HAVE_EXP2_BUILTIN 1
#  endif
#endif

#define AS1 __attribute__((address_space(1)))
#define AS3 __attribute__((address_space(3)))

typedef __attribute__((ext_vector_type(4))) int v4i;
typedef AS1 v4i* gv4i_ptr;
typedef AS3 v4i* lv4i_ptr;

__device__ __forceinline__ void async_ld16(const void* gsrc, void* ldst) {
#if defined(HAVE_ASYNC_BUILTIN)
  __builtin_amdgcn_global_load_async_to_lds_b128(
      (gv4i_ptr)gsrc, (lv4i_ptr)ldst, 0, 0);
#else
  unsigned loff = (unsigned)(unsigned long long)(AS3 char*)ldst;
  asm volatile("global_load_async_to_lds_b128 %0, %1, off"
               :: "v"(loff), "v"(gsrc) : "memory");
#endif
}

__device__ __forceinline__ void wait_async_le1() {
#if defined(HAVE_WAITASYNC_BUILTIN)
  asm volatile("" ::: "memory");
  __builtin_amdgcn_s_wait_asynccnt(1);
  asm volatile("" ::: "memory");
#else
  asm volatile("s_wait_asynccnt 0x1" ::: "memory");
#endif
}

__device__ __forceinline__ void wait_async_le0() {
#if defined(HAVE_WAITASYNC_BUILTIN)
  asm volatile("" ::: "memory");
  __builtin_amdgcn_s_wait_asynccnt(0);
  asm volatile("" ::: "memory");
#else
  asm volatile("s_wait_asynccnt 0x0" ::: "memory");
#endif
}

// Raw hardware exp2 (v_exp_f32), no libm range fixup: argument is already in
// log2 units and always <= 0; underflow flushes toward 0 just like the
// reference's fp32 exp underflow for negligible mixture tails.
__device__ __forceinline__ float hw_exp2(float x) {
#if defined(HAVE_EXP2_BUILTIN)
  return __builtin_amdgcn_exp2f(x);
#else
  return exp2f(x);
#endif
}

// Force a wave-uniform value into an SGPR.
__device__ __forceinline__ float rfl(float v) {
  return __int_as_float(__builtin_amdgcn_readfirstlane(__float_as_int(v)));
}

#define LOG2E 1.44269504088896340736f

// ---- per-component constants (run once, 16 lanes) --------------------------
// All log-domain quantities are pre-scaled by log2(e) so the main kernel can
// use the native v_exp_f32 (which is exp2) with no per-element argument scaling.
// prm[4l+0]=mean
// prm[4l+1]=log2e/(2var)        (dist scale, log2 units)
// prm[4l+2]=inv_norm
// prm[4l+3]=log2(inv_norm)      (hard-path bias, log2 units)
__global__ void gmmq_param_kernel(const float* __restrict__ mean,
                                  const float* __restrict__ log_std,
                                  const float* __restrict__ log_pi,
                                  float* __restrict__ prm) {
  int l = (int)threadIdx.x;
  if (l >= LCOMP) return;
  float mn = log_pi[0];
  for (int j = 1; j < LCOMP; ++j) mn = fminf(mn, log_pi[j]);
  float s = 0.f;
  for (int j = 0; j < LCOMP; ++j) s += __expf(log_pi[j] - mn);
  float np   = __expf(log_pi[l] - mn) / s;              // normalized pi
  float sd   = __expf(log_std[l]);
  float var  = sd * sd + 1e-3f;                          // DIV
  float invn = np / sqrtf(2.0f * 3.14159265358979323846f * var);
  prm[4 * l + 0] = mean[l];
  prm[4 * l + 1] = LOG2E / (2.0f * var);
  prm[4 * l + 2] = invn;
  prm[4 * l + 3] = __log2f(invn);
}

// ---- per-element GMM quantize (dist in log2 units) -------------------------
__device__ __forceinline__ void quant1(float xv,
                                       const float* cm, const float* ci,
                                       const float* cn, const float* cg,
                                       float& qo, int& so) {
  float dist[LCOMP];
#pragma unroll
  for (int l = 0; l < LCOMP; ++l) {
    float d = xv - cm[l];
    dist[l] = d * d * ci[l];
  }
  // soft softmax: z = -dist, max z = -min dist
  float mind = dist[0];
#pragma unroll
  for (int l = 1; l < LCOMP; ++l) mind = fminf(mind, dist[l]);
  float sum = 0.f, acc = 0.f;
#pragma unroll
  for (int l = 0; l < LCOMP; ++l) {
    float e = hw_exp2(mind - dist[l]) * cn[l];  // bare v_exp_f32
    sum += e;
    acc = fmaf(e, cm[l], acc);
  }
  float softout = __fdividef(acc, sum);  // sum in (0, 16*max(inv_norm)], safe
  // hard path, log2 domain: argmax(log2(inv_norm) - 1e4*dist2) == argmax(phi_hard)
  // (everything uniformly scaled by log2e, argmax invariant)
  float best  = fmaf(dist[0], -1e4f, cg[0]);
  float bmean = cm[0];
  int   bi    = 0;
#pragma unroll
  for (int l = 1; l < LCOMP; ++l) {
    float sc = fmaf(dist[l], -1e4f, cg[l]);
    bool g   = sc > best;
    best  = g ? sc : best;
    bmean = g ? cm[l] : bmean;
    bi    = g ? l : bi;
  }
  qo = softout + (bmean - softout);  // straight-through forward value
  so = bi;
}

// ---- main streaming kernel: double-buffered async-to-LDS pipeline ----------
__global__ __launch_bounds__(TPB) void gmmq_main_kernel(
    const float* __restrict__ x, const float* __restrict__ prm,
    float* __restrict__ qout, int* __restrict__ sym, int n, int nvec) {
  __shared__ float4 lbuf[2][TPB];

  // 64 wave-uniform constants -> SGPRs
  float cm[LCOMP], ci[LCOMP], cn[LCOMP], cg[LCOMP];
#pragma unroll
  for (int l = 0; l < LCOMP; ++l) {
    cm[l] = rfl(prm[4 * l + 0]);
    ci[l] = rfl(prm[4 * l + 1]);
    cn[l] = rfl(prm[4 * l + 2]);
    cg[l] = rfl(prm[4 * l + 3]);
  }

  const int tid    = (int)threadIdx.x;
  const int stride = (int)gridDim.x * TPB;
  const float4* x4 = (const float4*)x;

  // prologue prefetch into stage 0 (clamped address: always exactly 1 async
  // per wave per stage, so s_wait_asynccnt counts are deterministic)
  if (nvec > 0) {
    int i0 = (int)blockIdx.x * TPB + tid;
    if (i0 >= nvec) i0 = nvec - 1;
    async_ld16(x4 + i0, &lbuf[0][tid]);
  }

  int cur = 0;
  for (int base = (int)blockIdx.x * TPB; base < nvec; base += stride) {
    const int nbase = base + stride;   // uniform across block
    if (nbase < nvec) {
      int ni = nbase + tid;
      if (ni >= nvec) ni = nvec - 1;
      async_ld16(x4 + ni, &lbuf[cur ^ 1][tid]);  // prefetch next stage
      wait_async_le1();                           // oldest (current) landed
    } else {
      wait_async_le0();
    }
    const int i = base + tid;
    if (i < nvec) {
      float4 v = lbuf[cur][tid];
      float q0, q1, q2, q3;
      int   s0, s1, s2, s3;
      quant1(v.x, cm, ci, cn, cg, q0, s0);
      quant1(v.y, cm, ci, cn, cg, q1, s1);
      quant1(v.z, cm, ci, cn, cg, q2, s2);
      quant1(v.w, cm, ci, cn, cg, q3, s3);
      ((float4*)qout)[i] = make_float4(q0, q1, q2, q3);
      ((int4*)sym)[i]    = make_int4(s0, s1, s2, s3);
    }
    cur ^= 1;
  }

  // scalar tail (n % 4 leftovers)
  if (blockIdx.x == 0) {
    int t = nvec * 4 + tid;
    if (t < n) {
      float q;
      int s;
      quant1(x[t], cm, ci, cn, cg, q, s);
      qout[t] = q;
      sym[t]  = s;
    }
  }
}

extern "C" void kernel_launch(void* const* d_in, const int* in_sizes, int n_in,
                              void* d_out, int out_size, void* d_ws, size_t ws_size,
                              hipStream_t stream) {
  (void)n_in; (void)out_size; (void)ws_size;
  const float* x       = (const float*)d_in[0];
  const float* mean    = (const float*)d_in[1];
  const float* log_std = (const float*)d_in[2];
  const float* log_pi  = (const float*)d_in[3];
  const int n = in_sizes[0];

  float* prm = (float*)d_ws;           // 64 floats of scratch
  float* q   = (float*)d_out;          // N floats
  int*   sym = (int*)d_out + n;        // N int32 symbols

  gmmq_param_kernel<<<1, 32, 0, stream>>>(mean, log_std, log_pi, prm);

  int nvec   = n / 4;
  int blocks = (nvec + TPB - 1) / TPB;
  if (blocks > MAXBLOCKS) blocks = MAXBLOCKS;
  if (blocks < 1) blocks = 1;
  gmmq_main_kernel<<<blocks, TPB, 0, stream>>>(x, prm, q, sym, n, nvec);
}